// NodeReconstructionModule_36636071035262
// MI455X (gfx1250) — compile-verified
//
#include <hip/hip_runtime.h>

#define NN 100000
#define EE 600000
#define KD 128   // IN == H == 128

typedef __attribute__((ext_vector_type(2))) float v2f;
typedef __attribute__((ext_vector_type(8))) float v8f;

// ---------------- utility kernels ----------------

__global__ void zero_kernel(float* __restrict__ p, long n) {
  long i = (long)blockIdx.x * blockDim.x + threadIdx.x;
  long stride = (long)gridDim.x * blockDim.x;
  for (; i < n; i += stride) p[i] = 0.0f;
}

__global__ void degree_kernel(const int* __restrict__ src, const int* __restrict__ dst,
                              float* __restrict__ deg_out, float* __restrict__ deg_in, int E) {
  int i = blockIdx.x * blockDim.x + threadIdx.x;
  if (i < E) {
    atomicAdd(&deg_out[src[i]], 1.0f);
    atomicAdd(&deg_in[dst[i]], 1.0f);
  }
}

__global__ void rsqrt_kernel(float* __restrict__ deg, int n) {
  int i = blockIdx.x * blockDim.x + threadIdx.x;
  if (i < n) deg[i] = rsqrtf(fmaxf(deg[i], 1.0f));
}

// wave-per-edge: gather feat[src]*rsqrt(outdeg[src]) (float4/lane), scatter-add to agg[dst]
__global__ void scatter_kernel(const float* __restrict__ feat, const int* __restrict__ src,
                               const int* __restrict__ dst, const float* __restrict__ sc_out,
                               float* __restrict__ agg, int E) {
  int wave = (int)((blockIdx.x * (long)blockDim.x + threadIdx.x) >> 5);
  int lane = threadIdx.x & 31;
  if (wave >= E) return;
  int s = src[wave];
  int d = dst[wave];
  float sc = sc_out[s];
  float4 v = ((const float4*)(feat + (long)s * KD))[lane];
  float* o = agg + (long)d * KD + lane * 4;
  atomicAdd(o + 0, v.x * sc);
  atomicAdd(o + 1, v.y * sc);
  atomicAdd(o + 2, v.z * sc);
  atomicAdd(o + 3, v.w * sc);
}

// ---------------- WMMA f32 GEMM ----------------
// out[M,128] = act( (sum_{s<NSRC} As * rowscale) @ W[128,128] + bias )
// One wave owns a 16x128 output row-block: A fragment loaded once per K-step,
// 8 independent V_WMMA_F32_16X16X4_F32 against 8 B fragments from LDS.
// W is staged in LDS K-pair-swizzled so each B fragment (W[ka][col], W[ka+1][col])
// is one contiguous 8-byte ds_load_b64 — no register re-pairing movs.
//   Ws2[p*256 + col*2 + r] = W[(2p + r)*128 + col],  p = 0..63, r = 0..1
template <int NSRC, bool USE_SCALE, bool DO_RELU>
__global__ __launch_bounds__(256) void gemm_wmma_kernel(
    const float* __restrict__ A0, const float* __restrict__ A1,
    const float* __restrict__ A2, const float* __restrict__ rowscale,
    const float* __restrict__ W, const float* __restrict__ bias,
    float* __restrict__ out, int M) {
  __shared__ float Ws2[(KD / 2) * (KD * 2)];  // 64 KB, swizzled
  for (int i = threadIdx.x; i < (KD / 2) * KD; i += blockDim.x) {
    int p = i >> 7;        // K-pair index
    int col = i & 127;     // column
    float2 w;
    w.x = W[(2 * p) * KD + col];
    w.y = W[(2 * p + 1) * KD + col];
    *(float2*)&Ws2[p * (KD * 2) + col * 2] = w;
  }
  __syncthreads();

  const int wid  = threadIdx.x >> 5;
  const int lane = threadIdx.x & 31;
  const int mt   = blockIdx.x * 8 + wid;  // one 16-row block per wave
  if (mt * 16 >= M) return;

  const int half = lane >> 4;    // 0: K={0,1}, M+=0 ; 1: K={2,3}, M+=8
  const int l15  = lane & 15;
  const int arow = mt * 16 + l15;
  const float rs = USE_SCALE ? rowscale[arow] : 1.0f;
  const float* Ar0 = A0 + (long)arow * KD;
  const float* Ar1 = A1 + (long)arow * KD;
  const float* Ar2 = A2 + (long)arow * KD;

  v8f c[8];
  #pragma unroll
  for (int n = 0; n < 8; ++n) c[n] = (v8f){};

  for (int k = 0; k < KD; k += 4) {
    const int ka = k + half * 2;          // even
    const int p  = ka >> 1;               // K-pair for this half
    float2 av = *(const float2*)(Ar0 + ka);   // 8B aligned
    if (NSRC >= 2) {
      float2 a1 = *(const float2*)(Ar1 + ka);
      av.x += a1.x; av.y += a1.y;
    }
    if (NSRC >= 3) {
      float2 a2 = *(const float2*)(Ar2 + ka);
      av.x += a2.x; av.y += a2.y;
    }
    v2f a;
    if (USE_SCALE) { a[0] = av.x * rs; a[1] = av.y * rs; }
    else           { a[0] = av.x;      a[1] = av.y;      }
    const float* brow = &Ws2[p * (KD * 2) + l15 * 2];
    #pragma unroll
    for (int n = 0; n < 8; ++n) {
      v2f b = *(const v2f*)(brow + n * 32);  // ds_load_b64, offsets 0..896 B
      // 8 args: (neg_a, A, neg_b, B, c_mod, C, reuse_a, reuse_b)
      c[n] = __builtin_amdgcn_wmma_f32_16x16x4_f32(false, a, false, b, (short)0, c[n],
                                                   false, false);
    }
  }

  // C layout: VGPR j -> M = mt*16 + j + (lane>=16 ? 8 : 0); N = n*16 + (lane&15)
  float* orow = out + ((long)(mt * 16 + half * 8)) * KD;
  #pragma unroll
  for (int n = 0; n < 8; ++n) {
    const int col = n * 16 + l15;
    const float bi = bias[col];
    #pragma unroll
    for (int j = 0; j < 8; ++j) {
      float v = c[n][j] + bi;
      if (DO_RELU) v = fmaxf(v, 0.0f);
      orow[(long)j * KD + col] = v;
    }
  }
}

// ---------------- host launch ----------------

extern "C" void kernel_launch(void* const* d_in, const int* in_sizes, int n_in,
                              void* d_out, int out_size, void* d_ws, size_t ws_size,
                              hipStream_t stream) {
  const float* feat = (const float*)d_in[0];
  const int* src[3] = {(const int*)d_in[1], (const int*)d_in[3], (const int*)d_in[5]};
  const int* dst[3] = {(const int*)d_in[2], (const int*)d_in[4], (const int*)d_in[6]};
  const float* We[3] = {(const float*)d_in[7], (const float*)d_in[9], (const float*)d_in[11]};
  const float* be[3] = {(const float*)d_in[8], (const float*)d_in[10], (const float*)d_in[12]};
  const float* Wr1 = (const float*)d_in[13];
  const float* br1 = (const float*)d_in[14];
  const float* Wr2 = (const float*)d_in[15];
  const float* br2 = (const float*)d_in[16];

  float* out = (float*)d_out;
  float* recon = out;                              // output 0
  float* h[3] = {out + (long)NN * KD,              // h0
                 out + 2L * NN * KD,               // h1
                 out + 3L * NN * KD};              // h2

  float* ws   = (float*)d_ws;
  float* agg  = ws;                    // N*128 floats
  float* tbuf = ws + (long)NN * KD;    // N*128 floats
  float* deg  = ws + 2L * NN * KD;     // 2*N floats: [0,N)=out-deg, [N,2N)=in-deg

  const int gemm_blocks = (NN / 16 + 7) / 8;       // 782: one wave per 16-row block

  for (int i = 0; i < 3; ++i) {
    zero_kernel<<<2048, 256, 0, stream>>>(agg, (long)NN * KD);
    zero_kernel<<<(2 * NN + 255) / 256, 256, 0, stream>>>(deg, 2L * NN);
    degree_kernel<<<(EE + 255) / 256, 256, 0, stream>>>(src[i], dst[i], deg, deg + NN, EE);
    rsqrt_kernel<<<(2 * NN + 255) / 256, 256, 0, stream>>>(deg, 2 * NN);
    scatter_kernel<<<(EE + 7) / 8, 256, 0, stream>>>(feat, src[i], dst[i], deg, agg, EE);
    // h_i = relu((agg * rsqrt(in_deg)) @ We_i + be_i)
    gemm_wmma_kernel<1, true, true><<<gemm_blocks, 256, 0, stream>>>(
        agg, agg, agg, deg + NN, We[i], be[i], h[i], NN);
  }

  // t = relu((h0+h1+h2) @ Wr1 + br1)   (combine fused into A-fragment load)
  gemm_wmma_kernel<3, false, true><<<gemm_blocks, 256, 0, stream>>>(
      h[0], h[1], h[2], deg, Wr1, br1, tbuf, NN);
  // reconstructed = t @ Wr2 + br2
  gemm_wmma_kernel<1, false, false><<<gemm_blocks, 256, 0, stream>>>(
      tbuf, tbuf, tbuf, deg, Wr2, br2, recon, NN);
}